// SwemCat_20572893348703
// MI455X (gfx1250) — compile-verified
//
#include <hip/hip_runtime.h>
#include <hip/hip_bf16.h>

// SWEM pooling: gather + masked max/mean over token embeddings.
// Memory-bound (table fits in 192MB L2). No matmul structure -> no WMMA;
// CDNA5 paths used: global_load_async_to_lds_b32 / s_wait_asynccnt for index
// staging, ds_load_b128 broadcast reads of staged indices, global_prefetch
// (via __builtin_prefetch) for row warming.

#define TT 30
#define TD 150
#define TD_BASE 32          // desc indices staged at word 32 -> 16B-aligned quads
#define ED 512
#define NEGV (-1e30f)

typedef int v4i __attribute__((ext_vector_type(4)));

struct Acc4 {
    float4 mx;
    float4 sm;
};

__device__ __forceinline__ void acc_row(Acc4& a, const float4 v) {
    a.mx.x = fmaxf(a.mx.x, v.x); a.mx.y = fmaxf(a.mx.y, v.y);
    a.mx.z = fmaxf(a.mx.z, v.z); a.mx.w = fmaxf(a.mx.w, v.w);
    a.sm.x += v.x; a.sm.y += v.y; a.sm.z += v.z; a.sm.w += v.w;
}

__device__ __forceinline__ const float4* row_ptr(const float* __restrict__ w2v,
                                                 int idx, int tid) {
    return (const float4*)(w2v + (size_t)idx * ED + (size_t)tid * 4);
}

// Explicit LDS reads: the LDS data is produced by async-DMA inline asm, so
// read it back with real ds_load_* (wait dscnt inside the asm since the
// compiler cannot track the counter for asm outputs).
__device__ __forceinline__ v4i lds_load_q(unsigned byte_off) {
    v4i q;
    asm volatile("ds_load_b128 %0, %1\n\t"
                 "s_wait_dscnt 0x0"
                 : "=v"(q) : "v"(byte_off));
    return q;
}

__device__ __forceinline__ int lds_load_1(unsigned byte_off) {
    int r;
    asm volatile("ds_load_b32 %0, %1\n\t"
                 "s_wait_dscnt 0x0"
                 : "=v"(r) : "v"(byte_off));
    return r;
}

// Pool over `len` tokens whose indices live in LDS at byte offset lds_base.
// 1-stage pipeline: load next index quad + prefetch its rows, while
// gathering/accumulating the current quad's rows.
__device__ __forceinline__ Acc4 pool(unsigned lds_base, int len,
                                     const float* __restrict__ w2v, int tid) {
    Acc4 a;
    a.mx = make_float4(NEGV, NEGV, NEGV, NEGV);
    a.sm = make_float4(0.f, 0.f, 0.f, 0.f);
    int t = 0;
    if (len >= 4) {
        v4i cur = lds_load_q(lds_base);
        for (; t + 8 <= len; t += 4) {
            const v4i nxt = lds_load_q(lds_base + 4u * (unsigned)(t + 4));
            // Warm L2 one quad ahead (global_prefetch_b8).
            __builtin_prefetch((const void*)row_ptr(w2v, nxt[0], tid), 0, 0);
            __builtin_prefetch((const void*)row_ptr(w2v, nxt[1], tid), 0, 0);
            __builtin_prefetch((const void*)row_ptr(w2v, nxt[2], tid), 0, 0);
            __builtin_prefetch((const void*)row_ptr(w2v, nxt[3], tid), 0, 0);
            const float4 v0 = *row_ptr(w2v, cur[0], tid);
            const float4 v1 = *row_ptr(w2v, cur[1], tid);
            const float4 v2 = *row_ptr(w2v, cur[2], tid);
            const float4 v3 = *row_ptr(w2v, cur[3], tid);
            acc_row(a, v0); acc_row(a, v1); acc_row(a, v2); acc_row(a, v3);
            cur = nxt;
        }
        // Last full quad (t + 4 <= len < t + 8).
        {
            const float4 v0 = *row_ptr(w2v, cur[0], tid);
            const float4 v1 = *row_ptr(w2v, cur[1], tid);
            const float4 v2 = *row_ptr(w2v, cur[2], tid);
            const float4 v3 = *row_ptr(w2v, cur[3], tid);
            acc_row(a, v0); acc_row(a, v1); acc_row(a, v2); acc_row(a, v3);
            t += 4;
        }
    }
    for (; t < len; ++t) {
        const int i = lds_load_1(lds_base + 4u * (unsigned)t);
        const float4 v = *row_ptr(w2v, i, tid);
        acc_row(a, v);
    }
    return a;
}

__device__ __forceinline__ void store_pool(float* __restrict__ out_max,
                                           float* __restrict__ out_avg,
                                           Acc4 a, int len) {
    const float denom = (float)(len > 0 ? len : 1);
    float4 mx = a.mx;
    if (len == 0) mx = make_float4(0.f, 0.f, 0.f, 0.f);
    float4 av;
    av.x = a.sm.x / denom; av.y = a.sm.y / denom;
    av.z = a.sm.z / denom; av.w = a.sm.w / denom;
    *(float4*)out_max = mx;
    *(float4*)out_avg = av;
}

__global__ __launch_bounds__(128, 8)
void swem_pool_kernel(const int* __restrict__ title,
                      const int* __restrict__ desc,
                      const int* __restrict__ t_len,
                      const int* __restrict__ d_len,
                      const float* __restrict__ w2v,
                      float* __restrict__ out) {
    __shared__ int sIdx[TD_BASE + TD];   // only LDS object -> group offset 0

    const int b   = blockIdx.x;
    const int tid = threadIdx.x;

    // ---- Stage the 180 token indices into LDS with CDNA5 async copies ----
    {
        // Escape the LDS object's address into the asm so the "memory"
        // clobber is known to alias it.
        const unsigned long long lds_base = (unsigned long long)(uintptr_t)sIdx;
        const int total = TT + TD;
        for (int base = 0; base < total; base += 128) {
            const int i = base + tid;
            if (i < total) {
                const int* gsrc = (i < TT) ? (title + (size_t)b * TT + i)
                                           : (desc  + (size_t)b * TD + (i - TT));
                const unsigned lds_off =
                    (i < TT) ? (unsigned)(4 * i)
                             : (unsigned)(4 * (TD_BASE + (i - TT)));
                const unsigned long long gaddr = (unsigned long long)gsrc;
                asm volatile("global_load_async_to_lds_b32 %0, %1, off"
                             :: "v"(lds_off), "v"(gaddr), "s"(lds_base)
                             : "memory");
            }
        }
        asm volatile("s_wait_asynccnt 0x0" ::: "memory");
        __syncthreads();
    }

    const int tl = t_len[b];   // block-uniform -> scalar loads
    const int dl = d_len[b];

    float* ob = out + (size_t)b * (4 * ED) + (size_t)tid * 4;

    // Title pool -> slots 0 (max) and 2 (avg)
    {
        Acc4 a = pool(0u, tl, w2v, tid);
        store_pool(ob + 0 * ED, ob + 2 * ED, a, tl);
    }
    // Desc pool -> slots 1 (max) and 3 (avg)
    {
        Acc4 a = pool((unsigned)(4 * TD_BASE), dl, w2v, tid);
        store_pool(ob + 1 * ED, ob + 3 * ED, a, dl);
    }
}

extern "C" void kernel_launch(void* const* d_in, const int* in_sizes, int n_in,
                              void* d_out, int out_size, void* d_ws, size_t ws_size,
                              hipStream_t stream) {
    const int*   title = (const int*)d_in[0];
    const int*   desc  = (const int*)d_in[1];
    const int*   tlen  = (const int*)d_in[2];
    const int*   dlen  = (const int*)d_in[3];
    // d_in[4] = mode (unused by reference)
    const float* w2v   = (const float*)d_in[5];
    float*       out   = (float*)d_out;

    const int B = in_sizes[2];          // 2048 rows (t_len element count)
    swem_pool_kernel<<<B, 128, 0, stream>>>(title, desc, tlen, dlen, w2v, out);
    (void)n_in; (void)out_size; (void)d_ws; (void)ws_size;
}